// Rcell_9990093931074
// MI455X (gfx1250) — compile-verified
//
#include <hip/hip_runtime.h>
#include <stdint.h>

// Streaming, bandwidth-bound batched 2x2 R-cell update for gfx1250 (MI455X).
//
// Roofline: 64 B and ~140 FLOP per element -> 2.2 FLOP/B; at 23.3 TB/s HBM the
// kernel needs only ~51 TFLOP/s FP32, far below VALU peak -> pure HBM stream.
// WMMA is deliberately NOT used: the only shared-matrix terms are 2x2 GEMMs
// (6% utilization of a 16x16x4 tile) and would need cross-lane permutes that
// cost more than the FMAs they replace.
//
// Data movement (the resource that matters):
//  - inputs staged with GLOBAL_LOAD_ASYNC_TO_LDS_B128 (ASYNCcnt path, NT),
//    wave-private staging -> only s_wait_asynccnt, no barrier
//  - LDS layout SoA-by-16B-chunk -> every async write / ds_load_b128 has a
//    16 B lane stride = zero bank conflicts (64 banks x 4 B)
//  - outputs written with non-temporal global_store_b128 (256 MB stream,
//    touched once, > 192 MB L2 -> NT avoids L2 thrash)
//  - C/A are wave-uniform -> s_load_b128 scalar fetches

typedef float v4f __attribute__((ext_vector_type(4)));

#define DT_C   0.001f
#define MAXU_C 0.1f
#define BLOCK  256

__device__ __forceinline__ float clipf(float v, float lo, float hi) {
    return fminf(fmaxf(v, lo), hi);
}

// Async copy of 16 bytes global -> LDS for this lane (ASYNCcnt-tracked).
// vdst = 32-bit LDS byte offset (low 32 bits of flat LDS pointer),
// vaddr = 64-bit global address pair.
__device__ __forceinline__ void async_ld_b128(void* lds_ptr, const void* gptr) {
    unsigned loff = (unsigned)(uintptr_t)lds_ptr;
    unsigned long long ga = (unsigned long long)(uintptr_t)gptr;
    asm volatile("global_load_async_to_lds_b128 %0, %1, off th:TH_LOAD_NT"
                 :: "v"(loff), "v"(ga)
                 : "memory");
}

__device__ __forceinline__ void wait_async0() {
    asm volatile("s_wait_asynccnt 0x0" ::: "memory");
}

struct El {
    float o0, o1;
    float x0, x1;
    float n00, n01, n10, n11;
};

__device__ __forceinline__ El rcell_one(
    float d0, float d1,
    float s0, float s1,
    float v00, float v01, float v10, float v11,
    float c00, float c01, float c10, float c11,
    float a00, float a01, float a10, float a11)
{
    El r;
    // output = (C @ sts) * dt
    r.o0 = (c00 * s0 + c01 * s1) * DT_C;
    r.o1 = (c10 * s0 + c11 * s1) * DT_C;

    // xicov = cov @ C^T
    float x00 = v00 * c00 + v01 * c01;
    float x01 = v00 * c10 + v01 * c11;
    float x10 = v10 * c00 + v11 * c01;
    float x11 = v10 * c10 + v11 * c11;

    // M = A - xicov @ C
    float m00 = a00 - (x00 * c00 + x01 * c10);
    float m01 = a01 - (x00 * c01 + x01 * c11);
    float m10 = a10 - (x10 * c00 + x11 * c10);
    float m11 = a11 - (x10 * c01 + x11 * c11);

    // dx = (M @ sts)*dt + xicov @ dy ; x = sts + clip(dx)
    float dx0 = (m00 * s0 + m01 * s1) * DT_C + (x00 * d0 + x01 * d1);
    float dx1 = (m10 * s0 + m11 * s1) * DT_C + (x10 * d0 + x11 * d1);
    r.x0 = s0 + clipf(dx0, -MAXU_C, MAXU_C);
    r.x1 = s1 + clipf(dx1, -MAXU_C, MAXU_C);

    // cov_dt = cov@A + A@cov - xicov@xicov^T ; new_cov = clip(cov + cov_dt*dt)
    float g00 = (v00 * a00 + v01 * a10) + (a00 * v00 + a01 * v10) - (x00 * x00 + x01 * x01);
    float g01 = (v00 * a01 + v01 * a11) + (a00 * v01 + a01 * v11) - (x00 * x10 + x01 * x11);
    float g10 = (v10 * a00 + v11 * a10) + (a10 * v00 + a11 * v10) - (x10 * x00 + x11 * x01);
    float g11 = (v10 * a01 + v11 * a11) + (a10 * v01 + a11 * v11) - (x10 * x10 + x11 * x11);

    r.n00 = clipf(v00 + g00 * DT_C, -1.0f, 1.0f);
    r.n01 = clipf(v01 + g01 * DT_C, -1.0f, 1.0f);
    r.n10 = clipf(v10 + g10 * DT_C, -1.0f, 1.0f);
    r.n11 = clipf(v11 + g11 * DT_C, -1.0f, 1.0f);
    return r;
}

// Hot kernel: handles Bv elements (Bv % 4 == 0), 4 per thread, async-staged.
__global__ __launch_bounds__(BLOCK) void rcell_vec(
    const float* __restrict__ dy,
    const float* __restrict__ sts,
    const float* __restrict__ cov,
    const float* __restrict__ Cm,
    const float* __restrict__ Am,
    float* __restrict__ out,
    float* __restrict__ xo,
    float* __restrict__ nc,
    int Bv)
{
    __shared__ v4f sdyA[BLOCK], sdyB[BLOCK];
    __shared__ v4f sstA[BLOCK], sstB[BLOCK];
    __shared__ v4f scv0[BLOCK], scv1[BLOCK], scv2[BLOCK], scv3[BLOCK];

    const float c00 = Cm[0], c01 = Cm[1], c10 = Cm[2], c11 = Cm[3];
    const float a00 = Am[0], a01 = Am[1], a10 = Am[2], a11 = Am[3];

    const int tid  = threadIdx.x;
    const int base = (blockIdx.x * BLOCK + tid) * 4;
    if (base >= Bv) return;

    const float* gdy = dy  + (size_t)base * 2;
    const float* gst = sts + (size_t)base * 2;
    const float* gcv = cov + (size_t)base * 4;

    // ---- async stage: 8 x 16B per lane, ASYNCcnt-tracked ----
    async_ld_b128(&sdyA[tid], gdy);
    async_ld_b128(&sdyB[tid], gdy + 4);
    async_ld_b128(&sstA[tid], gst);
    async_ld_b128(&sstB[tid], gst + 4);
    async_ld_b128(&scv0[tid], gcv);
    async_ld_b128(&scv1[tid], gcv + 4);
    async_ld_b128(&scv2[tid], gcv + 8);
    async_ld_b128(&scv3[tid], gcv + 12);
    wait_async0();              // wave-private staging: no barrier needed

    // ---- read back from LDS (ds_load_b128, conflict-free) ----
    v4f dA = sdyA[tid];
    v4f dB = sdyB[tid];
    v4f sA = sstA[tid];
    v4f sB = sstB[tid];
    v4f v0 = scv0[tid];
    v4f v1 = scv1[tid];
    v4f v2 = scv2[tid];
    v4f v3 = scv3[tid];

    El e0 = rcell_one(dA.x, dA.y, sA.x, sA.y, v0.x, v0.y, v0.z, v0.w,
                      c00, c01, c10, c11, a00, a01, a10, a11);
    El e1 = rcell_one(dA.z, dA.w, sA.z, sA.w, v1.x, v1.y, v1.z, v1.w,
                      c00, c01, c10, c11, a00, a01, a10, a11);
    El e2 = rcell_one(dB.x, dB.y, sB.x, sB.y, v2.x, v2.y, v2.z, v2.w,
                      c00, c01, c10, c11, a00, a01, a10, a11);
    El e3 = rcell_one(dB.z, dB.w, sB.z, sB.w, v3.x, v3.y, v3.z, v3.w,
                      c00, c01, c10, c11, a00, a01, a10, a11);

    v4f o01 = {e0.o0, e0.o1, e1.o0, e1.o1};
    v4f o23 = {e2.o0, e2.o1, e3.o0, e3.o1};
    v4f x01 = {e0.x0, e0.x1, e1.x0, e1.x1};
    v4f x23 = {e2.x0, e2.x1, e3.x0, e3.x1};
    v4f n0  = {e0.n00, e0.n01, e0.n10, e0.n11};
    v4f n1  = {e1.n00, e1.n01, e1.n10, e1.n11};
    v4f n2  = {e2.n00, e2.n01, e2.n10, e2.n11};
    v4f n3  = {e3.n00, e3.n01, e3.n10, e3.n11};

    v4f* outp = (v4f*)(out + (size_t)base * 2);
    v4f* xop  = (v4f*)(xo  + (size_t)base * 2);
    v4f* ncp  = (v4f*)(nc  + (size_t)base * 4);

    __builtin_nontemporal_store(o01, outp);
    __builtin_nontemporal_store(o23, outp + 1);
    __builtin_nontemporal_store(x01, xop);
    __builtin_nontemporal_store(x23, xop + 1);
    __builtin_nontemporal_store(n0, ncp);
    __builtin_nontemporal_store(n1, ncp + 1);
    __builtin_nontemporal_store(n2, ncp + 2);
    __builtin_nontemporal_store(n3, ncp + 3);
}

// Tail kernel: scalar, handles elements [Bv, B) (at most 3). Launched only
// when B % 4 != 0.
__global__ void rcell_tail(
    const float* __restrict__ dy,
    const float* __restrict__ sts,
    const float* __restrict__ cov,
    const float* __restrict__ Cm,
    const float* __restrict__ Am,
    float* __restrict__ out,
    float* __restrict__ xo,
    float* __restrict__ nc,
    int Bv, int B)
{
    const int b = Bv + (int)threadIdx.x;
    if (b >= B) return;

    const float c00 = Cm[0], c01 = Cm[1], c10 = Cm[2], c11 = Cm[3];
    const float a00 = Am[0], a01 = Am[1], a10 = Am[2], a11 = Am[3];

    El e = rcell_one(dy[(size_t)b * 2], dy[(size_t)b * 2 + 1],
                     sts[(size_t)b * 2], sts[(size_t)b * 2 + 1],
                     cov[(size_t)b * 4], cov[(size_t)b * 4 + 1],
                     cov[(size_t)b * 4 + 2], cov[(size_t)b * 4 + 3],
                     c00, c01, c10, c11, a00, a01, a10, a11);
    out[(size_t)b * 2]     = e.o0;
    out[(size_t)b * 2 + 1] = e.o1;
    xo[(size_t)b * 2]      = e.x0;
    xo[(size_t)b * 2 + 1]  = e.x1;
    nc[(size_t)b * 4]      = e.n00;
    nc[(size_t)b * 4 + 1]  = e.n01;
    nc[(size_t)b * 4 + 2]  = e.n10;
    nc[(size_t)b * 4 + 3]  = e.n11;
}

extern "C" void kernel_launch(void* const* d_in, const int* in_sizes, int n_in,
                              void* d_out, int out_size, void* d_ws, size_t ws_size,
                              hipStream_t stream) {
    const float* dy  = (const float*)d_in[0];
    const float* sts = (const float*)d_in[1];
    const float* cov = (const float*)d_in[2];
    const float* Cm  = (const float*)d_in[3];
    const float* Am  = (const float*)d_in[4];

    const int B  = in_sizes[0] / 2;   // dy is [B,2]
    const int Bv = B & ~3;            // vectorized portion

    float* out = (float*)d_out;            // [B,2]
    float* xo  = out + (size_t)2 * B;      // [B,2]
    float* nc  = out + (size_t)4 * B;      // [B,2,2]

    if (Bv > 0) {
        const int nthr   = Bv / 4;
        const int blocks = (nthr + BLOCK - 1) / BLOCK;
        rcell_vec<<<blocks, BLOCK, 0, stream>>>(dy, sts, cov, Cm, Am, out, xo, nc, Bv);
    }
    if (B != Bv) {
        rcell_tail<<<1, 4, 0, stream>>>(dy, sts, cov, Cm, Am, out, xo, nc, Bv, B);
    }
}